// GumbelVectorQuantizer_11940009083260
// MI455X (gfx1250) — compile-verified
//
#include <hip/hip_runtime.h>
#include <math.h>

// Problem constants (from reference): B=4, T=2048, F_IN=768, G=2, V=320, D=128
#define BSZ   4
#define TSZ   2048
#define FIN   768
#define GRP   2
#define VV    320
#define DD    128
#define NROW  (BSZ * TSZ)          // 8192 rows
#define NCOL  (GRP * VV)           // 640 logit columns
#define XQW   (GRP * DD)           // 256 output features per row
#define XQ_ELEMS ((size_t)NROW * XQW)  // 2,097,152

typedef __attribute__((ext_vector_type(2))) float v2f;
typedef __attribute__((ext_vector_type(8))) float v8f;

// Fused kernel: one block = one 16-row tile.
//   phase 1: logits(16x640) = x_tile @ W^T + b  via V_WMMA_F32_16X16X4_F32 -> LDS
//   phase 2: per (row,group): argmax (first-index tiebreak) + softmax probs
//            accumulated into block-local LDS, flushed to global accumulators
//   phase 3: gather xq rows from codebook using the argmax indices
__global__ __launch_bounds__(256) void gvq_fused(
    const float* __restrict__ x,        // (8192, 768)
    const float* __restrict__ W,        // (640, 768)
    const float* __restrict__ bias,     // (640,)
    const float* __restrict__ codebook, // (2, 320, 128)
    float* __restrict__ out,            // xq (8192, 256) at d_out[0]
    float* __restrict__ g_avg,          // (640,) softmax-prob accumulator
    float* __restrict__ g_hard)         // (640,) hard one-hot count accumulator
{
    __shared__ float lds_logits[16 * NCOL];  // 40 KB
    __shared__ float lds_acc[NCOL];
    __shared__ float lds_hard[NCOL];
    __shared__ int   lds_k[16 * GRP];

    const int tid  = threadIdx.x;
    const int wv   = tid >> 5;      // wave id 0..7 (wave32)
    const int lane = tid & 31;
    const int half = lane >> 4;     // 0: lanes 0-15, 1: lanes 16-31
    const int ln   = lane & 15;
    const int row0 = blockIdx.x * 16;

    for (int i = tid; i < NCOL; i += 256) { lds_acc[i] = 0.0f; lds_hard[i] = 0.0f; }

    // ---------------- phase 1: WMMA fp32 GEMM ----------------
    // A(16x4): lanes 0-15 hold K={0,1} in v0/v1; lanes 16-31 hold K={2,3}.
    // B(4x16): v0 lanes0-15 = row K=0, lanes16-31 = row K=2; v1 likewise K=1/K=3.
    // Both reduce to: lane reads its row at k-offset 2*half.
    const float* xrow = x + (size_t)(row0 + ln) * FIN;
    const int koff = half * 2;
    for (int nt = wv; nt < NCOL / 16; nt += 8) {   // 40 N-tiles, 5 per wave
        const float* wrow = W + (size_t)(nt * 16 + ln) * FIN;
        v8f c = {};
        #pragma unroll 4
        for (int kk = 0; kk < FIN; kk += 4) {
            v2f a = *(const v2f*)(xrow + kk + koff);
            v2f b = *(const v2f*)(wrow + kk + koff);
            // 8 args: (neg_a, A, neg_b, B, c_mod, C, reuse_a, reuse_b)
            c = __builtin_amdgcn_wmma_f32_16x16x4_f32(
                    false, a, false, b, (short)0, c, false, false);
        }
        const int   col = nt * 16 + ln;
        const float bv  = bias[col];
        // C/D layout: VGPR r, lane L -> M = r + 8*half, N = ln
        #pragma unroll
        for (int r = 0; r < 8; ++r)
            lds_logits[(r + 8 * half) * NCOL + col] = c[r] + bv;
    }
    __syncthreads();

    // ---------------- phase 2: argmax + softmax ----------------
    // 32 (row,group) tasks; each wave handles 4, lane-strided over 320 logits.
    for (int q = 0; q < 4; ++q) {
        const int task = wv * 4 + q;
        const int row  = task >> 1;
        const int g    = task & 1;
        const float* lr = lds_logits + row * NCOL + g * VV;

        float vals[10];
        float m = -INFINITY;
        int   mi = 0;
        #pragma unroll
        for (int j = 0; j < 10; ++j) {
            float v = lr[lane + 32 * j];
            vals[j] = v;
            if (v > m) { m = v; mi = lane + 32 * j; }   // strict > : first index wins
        }
        #pragma unroll
        for (int off = 16; off > 0; off >>= 1) {
            float om = __shfl_xor(m, off, 32);
            int   oi = __shfl_xor(mi, off, 32);
            if (om > m || (om == m && oi < mi)) { m = om; mi = oi; }
        }
        float s = 0.0f;
        #pragma unroll
        for (int j = 0; j < 10; ++j) { vals[j] = expf(vals[j] - m); s += vals[j]; }
        #pragma unroll
        for (int off = 16; off > 0; off >>= 1) s += __shfl_xor(s, off, 32);
        const float inv = 1.0f / s;
        #pragma unroll
        for (int j = 0; j < 10; ++j)
            atomicAdd(&lds_acc[g * VV + lane + 32 * j], vals[j] * inv);
        if (lane == 0) {
            lds_k[row * GRP + g] = mi;
            atomicAdd(&lds_hard[g * VV + mi], 1.0f);
        }
    }
    __syncthreads();

    // flush block-local accumulators (1280 global atomics per block)
    for (int i = tid; i < NCOL; i += 256) {
        atomicAdd(&g_avg[i],  lds_acc[i]);
        atomicAdd(&g_hard[i], lds_hard[i]);
    }

    // ---------------- phase 3: gather xq ----------------
    for (int i = tid; i < 16 * XQW; i += 256) {
        const int mrow = i >> 8;        // / 256
        const int j    = i & 255;
        const int g    = j >> 7;        // / 128
        const int dd   = j & 127;
        const int k    = lds_k[mrow * GRP + g];
        out[(size_t)(row0 + mrow) * XQW + j] = codebook[(g * VV + k) * DD + dd];
    }
}

// Finalize: perplexities + the g*v scalar, written after the xq block.
__global__ __launch_bounds__(NCOL) void gvq_finalize(
    const float* __restrict__ g_avg,
    const float* __restrict__ g_hard,
    float* __restrict__ out3)   // out3[0]=g*v, [1]=code_ppl, [2]=prob_ppl
{
    __shared__ float th[NCOL];
    __shared__ float tp[NCOL];
    const int i = threadIdx.x;
    const float invN = 1.0f / (float)NROW;
    const float pc = g_hard[i] * invN;
    const float pp = g_avg[i]  * invN;
    th[i] = pc * logf(pc + 1e-7f);
    tp[i] = pp * logf(pp + 1e-7f);
    __syncthreads();
    if (i == 0) {
        float cp = 0.0f, px = 0.0f;
        for (int g = 0; g < GRP; ++g) {
            float sh = 0.0f, sp = 0.0f;
            for (int v = 0; v < VV; ++v) { sh += th[g * VV + v]; sp += tp[g * VV + v]; }
            cp += expf(-sh);
            px += expf(-sp);
        }
        out3[0] = (float)(GRP * VV);
        out3[1] = cp;
        out3[2] = px;
    }
}

extern "C" void kernel_launch(void* const* d_in, const int* in_sizes, int n_in,
                              void* d_out, int out_size, void* d_ws, size_t ws_size,
                              hipStream_t stream) {
    const float* x    = (const float*)d_in[0];   // (4,2048,768)
    const float* W    = (const float*)d_in[1];   // (640,768)
    const float* bias = (const float*)d_in[2];   // (640,)
    const float* cb   = (const float*)d_in[3];   // (2,320,128)
    float* out    = (float*)d_out;
    float* g_avg  = (float*)d_ws;                // 640 floats
    float* g_hard = g_avg + NCOL;                // 640 floats

    // zero accumulators every launch (graph-capturable)
    hipMemsetAsync(d_ws, 0, 2 * NCOL * sizeof(float), stream);

    gvq_fused<<<NROW / 16, 256, 0, stream>>>(x, W, bias, cb, out, g_avg, g_hard);
    gvq_finalize<<<1, NCOL, 0, stream>>>(g_avg, g_hard, out + XQ_ELEMS);
}